// KNN_13692355740339
// MI455X (gfx1250) — compile-verified
//
#include <hip/hip_runtime.h>
#include <math.h>

typedef __attribute__((ext_vector_type(2))) float v2f;
typedef __attribute__((ext_vector_type(8))) float v8f;

#define Bd   4
#define Cd   3
#define Hd   256
#define Wd   512
#define PAD  4      // (SEARCH-1)/2
#define TILE 16

// dist[b,h,w] = sqrt( min_{|di|<=4,|dj|<=4} sum_c (tfm[b,c,h,w]-obs_pad[b,c,h+di,w+dj])^2 )
// via d^2 = |t|^2 - 2*D where D = t.o - 0.5*|o|^2 computed by V_WMMA_F32_16X16X4_F32
// (K=0..2: channels, K=3: -0.5 on the B side, |o|^2 on the A side).
// Zero padding is exact: a zeroed A row (incl. the |o|^2 slot) gives D=0 -> cand=|t|^2.
__global__ __launch_bounds__(256) void knn_min_dist_wmma(
    const float* __restrict__ tfm, const float* __restrict__ obs,
    float* __restrict__ out)
{
    const int lane        = threadIdx.x & 31;
    const int waveInBlock = threadIdx.x >> 5;
    const int waveId      = blockIdx.x * 8 + waveInBlock;

    const int tilesPerRow = Wd / TILE;                 // 32
    const int tileIdx     = waveId % tilesPerRow;
    const int h           = (waveId / tilesPerRow) % Hd;
    const int b           =  waveId / (tilesPerRow * Hd);
    const int w0          = tileIdx * TILE;

    const int n    = lane & 15;   // D column = tfm pixel within tile
    const int half = lane >> 4;   // 0 -> K=0,1 / M rows 0..7 ; 1 -> K=2,3 / M rows 8..15

    const int    w       = w0 + n;
    const size_t planeHW = (size_t)Hd * Wd;
    const size_t bBase   = ((size_t)b * Cd) * planeHW;
    const size_t baseT   = bBase + (size_t)h * Wd + w;

    const float t0 = tfm[baseT];
    const float t1 = tfm[baseT + planeHW];
    const float t2 = tfm[baseT + 2 * planeHW];
    const float tnorm = t0 * t0 + t1 * t1 + t2 * t2;

    // B-matrix operand (4x16): rows 0..2 = tfm channels, row 3 = -0.5
    v2f bmat;
    if (half == 0) { bmat.x = t0; bmat.y = t1;    }
    else           { bmat.x = t2; bmat.y = -0.5f; }

    // Band-extraction masks. D element i holds m = half*8 + i.
    // A-tile t covers obs col (w0 - 4 + 16*t + m)  ->  delta = m - n - 4 + 16*t.
    // t=0 valid: n <= m <= n+8 ; t=1 valid: m <= n-8.  (exactly 9 deltas/pixel)
    const float NEG = -3.0e38f;
    float msel[2][8];
#pragma unroll
    for (int i = 0; i < 8; ++i) {
        const int m = half * 8 + i;
        msel[0][i] = ((m >= n) && (m <= n + 8)) ? 0.0f : NEG;
        msel[1][i] = (m <= n - 8)               ? 0.0f : NEG;
    }

    // This lane's A-row (obs column) for each tile, clamped so loads are
    // always in-bounds and unconditional; padding is zeroed via okf.
    const int colRaw[2] = { w0 - PAD + n, w0 - PAD + TILE + n };

    float maxD = NEG;
    const v8f czero = {};

#pragma unroll
    for (int dr = 0; dr < 9; ++dr) {
        const int  r      = h + dr - PAD;
        const bool rowOk  = (r >= 0) && (r < Hd);
        const int  rClamp = min(max(r, 0), Hd - 1);
        const size_t rowBase = bBase + (size_t)rClamp * Wd;

#pragma unroll
        for (int t = 0; t < 2; ++t) {
            const int   c      = colRaw[t];
            const bool  ok     = rowOk && (c >= 0) && (c < Wd);
            const int   cClamp = min(max(c, 0), Wd - 1);
            const float okf    = ok ? 1.0f : 0.0f;

            const size_t baseO = rowBase + cClamp;
            const float a0 = obs[baseO]                * okf;
            const float a1 = obs[baseO + planeHW]      * okf;
            const float a2 = obs[baseO + 2 * planeHW]  * okf;
            const float onorm = a0 * a0 + a1 * a1 + a2 * a2;

            // A-matrix operand (16x4): rows = obs columns, K3 = |o|^2
            v2f amat;
            if (half == 0) { amat.x = a0; amat.y = a1;    }
            else           { amat.x = a2; amat.y = onorm; }

            // D[m][n] = dot3(obs_m, tfm_n) - 0.5*|obs_m|^2
            v8f d = __builtin_amdgcn_wmma_f32_16x16x4_f32(
                false, amat, false, bmat, (short)0, czero, false, false);

#pragma unroll
            for (int i = 0; i < 8; ++i) {
                maxD = fmaxf(maxD, d[i] + msel[t][i]);
            }
        }
    }

    // Pixel n's window rows are split across lanes n and n+16 -> merge halves.
    const float other = __shfl_xor(maxD, 16, 32);
    maxD = fmaxf(maxD, other);

    float d2 = tnorm - 2.0f * maxD;
    d2 = fmaxf(d2, 0.0f);

    if (half == 0) {
        out[(size_t)b * planeHW + (size_t)h * Wd + w] = sqrtf(d2);
    }
}

extern "C" void kernel_launch(void* const* d_in, const int* in_sizes, int n_in,
                              void* d_out, int out_size, void* d_ws, size_t ws_size,
                              hipStream_t stream) {
    const float* tfm = (const float*)d_in[0];
    const float* obs = (const float*)d_in[1];
    float*       out = (float*)d_out;

    // one wave per (b, h, 16-pixel tile): 4*256*32 = 32768 waves, 8 waves/block
    const int totalWaves = Bd * Hd * (Wd / TILE);
    const int blocks     = totalWaves / 8;
    hipLaunchKernelGGL(knn_min_dist_wmma, dim3(blocks), dim3(256), 0, stream,
                       tfm, obs, out);
}